// LocallyConnected2d_10557029613636
// MI455X (gfx1250) — compile-verified
//
#include <hip/hip_runtime.h>
#include <hip/hip_bf16.h>

// Locally-connected 2D conv (unshared weights), fp32, via V_WMMA_F32_16X16X4_F32.
// x: (16,32,64,64)  weights: (64,64,32,32,3,3)  bias: (32,64,64) flat-added
// after torch-style raw reshape: out_flat[b*131072 + g] = dot(b,g) + bias_flat[g],
// g = h*2048 + w*32 + o.
//
// Per (h,w): GEMM  M=16(batch) x N=32(oc) x K=288(c*3*3) as 72 WMMA K=4 steps
// over two N-tiles. One wave per location, 8 locations per 256-thread block.
// x patch window staged in LDS with batch-minor layout [q=c*3+kh][col][b(pad17)]
// so A-fragment ds loads are bank-conflict-free (lanes stride-1).

typedef __attribute__((ext_vector_type(2))) float v2f;
typedef __attribute__((ext_vector_type(8))) float v8f;

#define BATCH   16
#define IC      32
#define HW      64
#define OC      32
#define KS      3
#define KDIM    (IC * KS * KS)        // 288
#define QDIM    (IC * KS)             // 96  (q = c*3 + kh)
#define WTILE   8                     // w-locations per workgroup (1 per wave)
#define LDSCOLS (WTILE + 2)           // 10-wide padded column window
#define BPAD    17                    // padded batch stride (conflict-free stores)
#define LDS_ELEMS (QDIM * LDSCOLS * BPAD)          // 16320 floats = 65280 B
#define STAGE_N   (BATCH * IC * KS * LDSCOLS)      // 15360 real elements
#define PER_BATCH (OC * HW * HW)      // 131072 floats per batch in output

__global__ __launch_bounds__(256)
void lc2d_wmma_kernel(const float* __restrict__ x,
                      const float* __restrict__ wts,
                      const float* __restrict__ bias,
                      float* __restrict__ out) {
    __shared__ float smem[LDS_ELEMS];

    const int h    = blockIdx.y;          // output row
    const int wblk = blockIdx.x * WTILE;  // first w of this block

    // ---- Stage x patch window (rows h-1..h+1, cols wblk-1..wblk+8, all b,c).
    // Iterate in global-memory order (col fastest) for coalescing; write to
    // batch-minor LDS layout smem[(q*10 + col)*17 + b].
    for (int sidx = threadIdx.x; sidx < STAGE_N; sidx += 256) {
        int col = sidx % LDSCOLS;
        int t   = sidx / LDSCOLS;
        int kh  = t % KS;
        int t2  = t / KS;
        int c   = t2 % IC;
        int b   = t2 / IC;
        int gh  = h - 1 + kh;
        int gw  = wblk - 1 + col;
        float v = 0.0f;
        if ((unsigned)gh < (unsigned)HW && (unsigned)gw < (unsigned)HW)
            v = x[(((size_t)b * IC + c) * HW + gh) * HW + gw];
        smem[((c * KS + kh) * LDSCOLS + col) * BPAD + b] = v;
    }
    __syncthreads();

    // ---- Per-wave GEMM for location (h, w)
    const int wave  = threadIdx.x >> 5;   // 0..7 -> w offset in window
    const int lane  = threadIdx.x & 31;
    const int lhalf = lane >> 4;          // 0: K pair {r0,r0+1}, 1: {r0+2,r0+3}
    const int lm    = lane & 15;          // A: m index; B: n index (mod 16)
    const int w     = wblk + wave;

    // Weight block for this location: 32 rows (o) x 288 (c*3*3), contiguous.
    const float* Wp = wts + (size_t)(h * HW + w) * (OC * KDIM);
    // Per-lane row pointers matching the B-fragment striping:
    // lane holds B[k0][n] = W_hw[n*288 + k0], k0 = r0 + 2*lhalf.
    const float* lanerow0 = Wp + (size_t)lm * KDIM + 2 * lhalf;          // n = lm
    const float* lanerow1 = Wp + (size_t)(lm + 16) * KDIM + 2 * lhalf;   // n = lm+16

    v8f c0 = {};  // accumulates o = 0..15
    v8f c1 = {};  // accumulates o = 16..31

#pragma unroll 4
    for (int r0 = 0; r0 < KDIM; r0 += 4) {
        const int k0 = r0 + 2 * lhalf;

        // B fragments: two consecutive floats of this lane's weight row
        // (8B-aligned; line reused over 8 consecutive steps from L0 -> RT hint).
        v2f bf0 = *(const v2f*)(lanerow0 + r0);
        v2f bf1 = *(const v2f*)(lanerow1 + r0);

        // A fragment from LDS: r = q*3 + kw  ->  single div/mod by 3.
        int q  = k0 / 3;
        int kw = k0 - q * 3;
        int addr = (q * LDSCOLS + wave + kw) * BPAD + lm;   // lanes: stride 1
        v2f af;
        af.x = smem[addr];
        af.y = smem[addr + (kw == 2 ? (LDSCOLS - 2) * BPAD : BPAD)];

        c0 = __builtin_amdgcn_wmma_f32_16x16x4_f32(
                 false, af, false, bf0, (short)0, c0, false, false);
        c1 = __builtin_amdgcn_wmma_f32_16x16x4_f32(
                 false, af, false, bf1, (short)0, c1, false, false);
    }

    // ---- Bias + scatter into torch-reshaped layout.
    // D layout: lane<16: D[m=v][n=lane]; lane>=16: D[m=v+8][n=lane-16].
    const size_t obase = (size_t)h * (HW * OC) + (size_t)w * OC;  // g without o
    const float b0 = bias[obase + lm];
    const float b1 = bias[obase + 16 + lm];

#pragma unroll
    for (int v = 0; v < 8; ++v) {
        const int m = v + 8 * lhalf;
        __builtin_nontemporal_store(c0[v] + b0,
            out + (size_t)m * PER_BATCH + obase + lm);
        __builtin_nontemporal_store(c1[v] + b1,
            out + (size_t)m * PER_BATCH + obase + 16 + lm);
    }
}

extern "C" void kernel_launch(void* const* d_in, const int* in_sizes, int n_in,
                              void* d_out, int out_size, void* d_ws, size_t ws_size,
                              hipStream_t stream) {
    (void)in_sizes; (void)n_in; (void)d_ws; (void)ws_size; (void)out_size;
    const float* x    = (const float*)d_in[0];
    const float* wts  = (const float*)d_in[1];
    const float* bias = (const float*)d_in[2];
    float* out        = (float*)d_out;

    dim3 grid(HW / WTILE, HW);   // (8, 64): w-block x h
    dim3 block(256);             // 8 waves (wave32)
    lc2d_wmma_kernel<<<grid, block, 0, stream>>>(x, wts, bias, out);
}